// CompositeLoss_25288767439246
// MI455X (gfx1250) — compile-verified
//
#include <hip/hip_runtime.h>

// CompositeLoss = 0.7*MSE + 0.3*(1 - mean SSIM), win=3, C1=1e-4, C2=9e-4.
// Memory floor: 201 MB @ 23.3 TB/s ~ 8.6 us; VALU is the practical limiter,
// so: 2 rows/lane packed f32 (v_pk_* / VOPD), shuffles only on raw x,y
// (4 ds_bpermute/row), 81-scaled SSIM algebra (no /9), v_rcp_f32 instead of
// precise divide, first iteration peeled (branch-free hot loop), WMMA f32
// wave reduction on the matrix pipe.

typedef float v2f __attribute__((ext_vector_type(2)));
typedef float v8f __attribute__((ext_vector_type(8)));

// ---- packed float2 helpers (map to v_pk_add/mul/fma_f32 or VOPD pairs) ----
__device__ __forceinline__ float2 f2(float a, float b) { return make_float2(a, b); }
__device__ __forceinline__ float2 f2add(float2 a, float2 b) {
    return f2(a.x + b.x, a.y + b.y);
}
__device__ __forceinline__ float2 f2mul(float2 a, float2 b) {
    return f2(a.x * b.x, a.y * b.y);
}
__device__ __forceinline__ float2 f2fma(float2 a, float2 b, float2 c) {
    return f2(fmaf(a.x, b.x, c.x), fmaf(a.y, b.y, c.y));
}
__device__ __forceinline__ float2 f2fmas(float s, float2 b, float2 c) {
    return f2(fmaf(s, b.x, c.x), fmaf(s, b.y, c.y));
}
__device__ __forceinline__ float2 shfl_down2(float2 v, int d) {
    return f2(__shfl_down(v.x, d, 32), __shfl_down(v.y, d, 32));
}

// Full 32-lane sum using V_WMMA_F32_16X16X4_F32.
// A (16x4 f32, 2 VGPRs): lane l<16 -> A[l][0..1], lane l>=16 -> A[l-16][2..3].
// With B = all-ones (layout-agnostic) D[m][n] = v_m + v_{m+16}; summing the 8
// D VGPRs per thread + shfl_xor(16) yields the full wave sum in every lane.
__device__ __forceinline__ float wave_sum_wmma(float v) {
    v2f a; a[0] = v;    a[1] = 0.0f;
    v2f b; b[0] = 1.0f; b[1] = 1.0f;
    v8f c = {};
    v8f d = __builtin_amdgcn_wmma_f32_16x16x4_f32(
        false, a, false, b, (short)0, c, false, false);
    float t = d[0] + d[1] + d[2] + d[3] + d[4] + d[5] + d[6] + d[7];
    t += __shfl_xor(t, 16, 32);
    return t;
}

struct HPair { float2 hx, hy, hxx, hyy, hxy; };

// Horizontal stage for row pair k (rows 2k, 2k+1): NT loads + 3-tap filters.
// Shuffles only the raw x,y values; product filters derived locally.
__device__ __forceinline__ HPair hstage(const float* __restrict__ xp,
                                        const float* __restrict__ yp,
                                        int k, int lane, float2& mse2) {
    float2 xv = f2(__builtin_nontemporal_load(xp + k * 64 + lane),
                   __builtin_nontemporal_load(xp + k * 64 + 32 + lane));
    float2 yv = f2(__builtin_nontemporal_load(yp + k * 64 + lane),
                   __builtin_nontemporal_load(yp + k * 64 + 32 + lane));

    float2 dd = f2(xv.x - yv.x, xv.y - yv.y);
    mse2 = f2fma(dd, dd, mse2);

    float2 x1 = shfl_down2(xv, 1), x2 = shfl_down2(xv, 2);
    float2 y1 = shfl_down2(yv, 1), y2 = shfl_down2(yv, 2);

    HPair h;
    h.hx  = f2add(f2add(xv, x1), x2);
    h.hy  = f2add(f2add(yv, y1), y2);
    h.hxx = f2fma(xv, xv, f2fma(x1, x1, f2mul(x2, x2)));
    h.hyy = f2fma(yv, yv, f2fma(y1, y1, f2mul(y2, y2)));
    h.hxy = f2fma(xv, yv, f2fma(x1, y1, f2mul(x2, y2)));
    return h;
}

__global__ __launch_bounds__(96)
void ssim_mse_partial(const float* __restrict__ x, const float* __restrict__ y,
                      float2* __restrict__ partials) {
    const int img  = blockIdx.x;
    const int wid  = threadIdx.x >> 5;  // wave = channel (3 waves/block)
    const int lane = threadIdx.x & 31;  // lane = column (wave32)

    const size_t base = ((size_t)img * 3 + wid) * 1024;
    const float* __restrict__ xp = x + base;
    const float* __restrict__ yp = y + base;

    // 81-scaled constants: K1 = 81*C1, K2 = 81*C2, covn = 9/8
    const float  K1 = 0.0081f, K2 = 0.0729f;
    const float2 K1v = f2(K1, K1), K2v = f2(K2, K2);

    float2 mse2 = f2(0.f, 0.f), ssum2 = f2(0.f, 0.f);

    // ---- peeled k = 0: seeds rolling pair (rows 0,1), no windows yet ----
    HPair p = hstage(xp, yp, 0, lane, mse2);

    // ---- k = 1..15: branch-free body, emits window rows (2k-2, 2k-1) ----
    #pragma unroll 5
    for (int k = 1; k < 16; ++k) {
        HPair h = hstage(xp, yp, k, lane, mse2);

        // vertical 3-tap on pairs: s = p + (p.y, h.x) + h
        float2 sx  = f2add(f2add(p.hx,  f2(p.hx.y,  h.hx.x)),  h.hx);
        float2 sy  = f2add(f2add(p.hy,  f2(p.hy.y,  h.hy.x)),  h.hy);
        float2 sxx = f2add(f2add(p.hxx, f2(p.hxx.y, h.hxx.x)), h.hxx);
        float2 syy = f2add(f2add(p.hyy, f2(p.hyy.y, h.hyy.x)), h.hyy);
        float2 sxy = f2add(f2add(p.hxy, f2(p.hxy.y, h.hxy.x)), h.hxy);

        // 81-scaled SSIM (valid for lanes 0..29; 30,31 masked once at the end)
        float2 t1  = f2mul(sx, sy);             // sx*sy
        float2 t2  = f2mul(sx, sx);             // sx^2
        float2 t23 = f2fma(sy, sy, t2);         // sx^2 + sy^2
        float2 s4  = f2add(sxx, syy);
        float2 a1  = f2fmas(2.0f, t1, K1v);
        float2 b1  = f2add(t23, K1v);
        float2 a2  = f2fmas(20.25f,  sxy, f2fmas(-2.25f,  t1,  K2v));
        float2 b2  = f2fmas(10.125f, s4,  f2fmas(-1.125f, t23, K2v));
        float2 num = f2mul(a1, a2);
        float2 den = f2mul(b1, b2);
        float2 r   = f2(__builtin_amdgcn_rcpf(den.x),
                        __builtin_amdgcn_rcpf(den.y));
        ssum2 = f2fma(num, r, ssum2);

        p = h;
    }

    float mse  = mse2.x + mse2.y;
    float ssum = (lane < 30) ? (ssum2.x + ssum2.y) : 0.0f;  // single mask

    // All lanes converged (EXEC all-ones) -> WMMA wave reduction
    float mse_w = wave_sum_wmma(mse);
    float s_w   = wave_sum_wmma(ssum);

    __shared__ float2 wsum[3];
    if (lane == 0) wsum[wid] = f2(mse_w, s_w);
    __syncthreads();
    if (threadIdx.x == 0) {
        float m = wsum[0].x + wsum[1].x + wsum[2].x;
        float s = wsum[0].y + wsum[1].y + wsum[2].y;
        partials[img] = f2(m, s);
    }
}

__global__ __launch_bounds__(256)
void finalize_loss(const float2* __restrict__ partials, float* __restrict__ out,
                   int n) {
    float m = 0.0f, s = 0.0f;
    for (int i = threadIdx.x; i < n; i += 256) {
        float2 p = partials[i];
        m += p.x; s += p.y;
    }
    for (int o = 16; o > 0; o >>= 1) {
        m += __shfl_xor(m, o, 32);
        s += __shfl_xor(s, o, 32);
    }
    __shared__ float mbuf[8], sbuf[8];
    const int wid = threadIdx.x >> 5, lane = threadIdx.x & 31;
    if (lane == 0) { mbuf[wid] = m; sbuf[wid] = s; }
    __syncthreads();
    if (threadIdx.x == 0) {
        float M = 0.0f, S = 0.0f;
        #pragma unroll
        for (int i = 0; i < 8; ++i) { M += mbuf[i]; S += sbuf[i]; }
        float nf = (float)n;
        float mse_mean  = M / (nf * 3072.0f);   // 3*32*32 elems / image
        float ssim_mean = S / (nf * 2700.0f);   // 3*30*30 windows / image
        out[0] = 0.7f * mse_mean + 0.3f * (1.0f - ssim_mean);
    }
}

extern "C" void kernel_launch(void* const* d_in, const int* in_sizes, int n_in,
                              void* d_out, int out_size, void* d_ws, size_t ws_size,
                              hipStream_t stream) {
    const float* x = (const float*)d_in[0];
    const float* y = (const float*)d_in[1];
    float* out = (float*)d_out;

    const int total = in_sizes[0];            // B*3*32*32
    const int nImg  = total / (3 * 32 * 32);  // 8192

    float2* partials = (float2*)d_ws;         // nImg * 8 bytes (64 KB)

    ssim_mse_partial<<<nImg, 96, 0, stream>>>(x, y, partials);
    finalize_loss<<<1, 256, 0, stream>>>(partials, out, nImg);
}